// MarketGCN_13219909337481
// MI455X (gfx1250) — compile-verified
//
#include <hip/hip_runtime.h>
#include <hip/hip_bf16.h>

#define NN 100000
#define NE 6400000
#define IND 128
#define HID 16
#define OUTD 3
#define NTILES (NN / 16)   // 6250, exact

typedef __attribute__((ext_vector_type(2))) float v2f;
typedef __attribute__((ext_vector_type(8))) float v8f;

// ---------- degree / normalization ----------
__global__ __launch_bounds__(256) void k_init_deg(float* __restrict__ deg) {
    int i = blockIdx.x * 256 + threadIdx.x;
    if (i < NN) deg[i] = 1.0f;  // self-loop
}

__global__ __launch_bounds__(256) void k_count_deg(const long long* __restrict__ dst,
                                                   float* __restrict__ deg) {
    int e = blockIdx.x * 256 + threadIdx.x;
    if (e < NE) atomicAdd(&deg[(int)dst[e]], 1.0f);
}

__global__ __launch_bounds__(256) void k_rsqrt(float* __restrict__ deg) {
    int i = blockIdx.x * 256 + threadIdx.x;
    if (i < NN) deg[i] = rsqrtf(deg[i]);  // deg >= 1 always (self-loops)
}

// ---------- layer-1 GEMM: h1 = x @ W1, via V_WMMA_F32_16X16X4_F32 ----------
__global__ __launch_bounds__(256) void k_gemm1_wmma(const float* __restrict__ x,
                                                    const float* __restrict__ W1,
                                                    float* __restrict__ h1) {
    __shared__ float w[IND * HID];  // 8 KB of the 320 KB/WGP LDS
    for (int i = threadIdx.x; i < IND * HID; i += 256) w[i] = W1[i];
    __syncthreads();

    const int wave = threadIdx.x >> 5;
    const int lane = threadIdx.x & 31;
    const int tile = blockIdx.x * 8 + wave;
    if (tile >= NTILES) return;          // wave-uniform: EXEC stays all-1s for WMMA

    const int half = lane >> 4;          // 0: K pair {0,1}, 1: K pair {2,3}
    const int lid  = lane & 15;          // A: row M;  B: col N
    const int koff = half * 2;
    const float* __restrict__ xr = x + (size_t)(tile * 16 + lid) * IND;

    v8f c = {};
#pragma unroll
    for (int kk = 0; kk < IND; kk += 4) {
        v2f a, b;
        a.x = xr[kk + koff];
        a.y = xr[kk + koff + 1];
        b.x = w[(kk + koff) * HID + lid];
        b.y = w[(kk + koff + 1) * HID + lid];
        // D = A(16x4,f32) * B(4x16,f32) + C(16x16,f32)
        c = __builtin_amdgcn_wmma_f32_16x16x4_f32(
                /*neg_a=*/false, a, /*neg_b=*/false, b,
                /*c_mod=*/(short)0, c, /*reuse_a=*/false, /*reuse_b=*/false);
    }

    // C/D layout: VGPR v, lane L -> (M = v + 8*(L/16), N = L%16)
#pragma unroll
    for (int v = 0; v < 8; ++v) {
        int m = tile * 16 + v + 8 * half;
        h1[(size_t)m * HID + lid] = c[v];
    }
}

// ---------- layer-1 aggregation ----------
__global__ __launch_bounds__(256) void k_self1(const float* __restrict__ h1,
                                               const float* __restrict__ dis,
                                               float* __restrict__ agg1) {
    int idx = blockIdx.x * 256 + threadIdx.x;
    if (idx < NN * HID) {
        int i = idx >> 4;
        float d = dis[i];
        agg1[idx] = h1[idx] * d * d;     // self-loop contribution, also init
    }
}

__global__ __launch_bounds__(256) void k_scatter1(const long long* __restrict__ src,
                                                  const long long* __restrict__ dst,
                                                  const float* __restrict__ h1,
                                                  const float* __restrict__ dis,
                                                  float* __restrict__ agg1) {
    long long gid = (long long)blockIdx.x * 256 + threadIdx.x;
    if (gid >= (long long)NE * HID) return;
    int e = (int)(gid >> 4);
    int j = (int)(gid & 15);
    int s = (int)src[e];
    int d = (int)dst[e];
    float norm = dis[s] * dis[d];
    atomicAdd(&agg1[d * HID + j], h1[s * HID + j] * norm);
}

__global__ __launch_bounds__(256) void k_relu_bias1(float* __restrict__ agg1,
                                                    const float* __restrict__ b1) {
    int idx = blockIdx.x * 256 + threadIdx.x;
    if (idx < NN * HID) {
        float v = agg1[idx] + b1[idx & 15];
        agg1[idx] = v > 0.0f ? v : 0.0f;
    }
}

// ---------- layer-2 GEMM: h2 = relu1 @ W2  (16x3, VALU) ----------
__global__ __launch_bounds__(256) void k_gemm2(const float* __restrict__ hr,
                                               const float* __restrict__ W2,
                                               float* __restrict__ h2) {
    int i = blockIdx.x * 256 + threadIdx.x;
    if (i >= NN) return;
    const float* __restrict__ r = hr + (size_t)i * HID;
    float a0 = 0.f, a1 = 0.f, a2 = 0.f;
#pragma unroll
    for (int k = 0; k < HID; ++k) {
        float hv = r[k];
        a0 = fmaf(hv, W2[k * OUTD + 0], a0);
        a1 = fmaf(hv, W2[k * OUTD + 1], a1);
        a2 = fmaf(hv, W2[k * OUTD + 2], a2);
    }
    h2[i * OUTD + 0] = a0;
    h2[i * OUTD + 1] = a1;
    h2[i * OUTD + 2] = a2;
}

// ---------- layer-2 aggregation into d_out ----------
__global__ __launch_bounds__(256) void k_init_out(const float* __restrict__ h2,
                                                  const float* __restrict__ dis,
                                                  const float* __restrict__ b2,
                                                  float* __restrict__ out) {
    int idx = blockIdx.x * 256 + threadIdx.x;
    if (idx < NN * OUTD) {
        int i = idx / OUTD;
        int j = idx - i * OUTD;
        float d = dis[i];
        out[idx] = b2[j] + h2[idx] * d * d;   // bias + self-loop term
    }
}

__global__ __launch_bounds__(256) void k_scatter2(const long long* __restrict__ src,
                                                  const long long* __restrict__ dst,
                                                  const float* __restrict__ h2,
                                                  const float* __restrict__ dis,
                                                  float* __restrict__ out) {
    int e = blockIdx.x * 256 + threadIdx.x;
    if (e >= NE) return;
    int s = (int)src[e];
    int d = (int)dst[e];
    float norm = dis[s] * dis[d];
#pragma unroll
    for (int j = 0; j < OUTD; ++j)
        atomicAdd(&out[d * OUTD + j], h2[s * OUTD + j] * norm);
}

extern "C" void kernel_launch(void* const* d_in, const int* in_sizes, int n_in,
                              void* d_out, int out_size, void* d_ws, size_t ws_size,
                              hipStream_t stream) {
    (void)in_sizes; (void)n_in; (void)out_size; (void)ws_size;
    const float*     x   = (const float*)d_in[0];
    const long long* ei  = (const long long*)d_in[1];   // int64 [2, NE]
    const float*     W1  = (const float*)d_in[2];
    const float*     b1  = (const float*)d_in[3];
    const float*     W2  = (const float*)d_in[4];
    const float*     b2  = (const float*)d_in[5];
    const long long* src = ei;
    const long long* dst = ei + NE;
    float* out = (float*)d_out;

    // workspace layout (float offsets, 128B-aligned)
    float* ws   = (float*)d_ws;
    float* dis  = ws;                 // 100000  (deg -> rsqrt in place)
    float* h1   = ws + 100032;        // 1600000
    float* agg1 = ws + 1700032;       // 1600000
    float* h2   = ws + 3300032;       // 300000   => ~14.4 MB total

    const int B = 256;
    const int gN    = (NN + B - 1) / B;
    const int gE    = (NE + B - 1) / B;
    const int gNH   = (NN * HID + B - 1) / B;
    const int gEH   = (int)(((long long)NE * HID + B - 1) / B);
    const int gNO   = (NN * OUTD + B - 1) / B;
    const int gTile = (NTILES + 7) / 8;   // 8 waves/block, 1 tile/wave

    k_init_deg  <<<gN,    B, 0, stream>>>(dis);
    k_count_deg <<<gE,    B, 0, stream>>>(dst, dis);
    k_rsqrt     <<<gN,    B, 0, stream>>>(dis);
    k_gemm1_wmma<<<gTile, B, 0, stream>>>(x, W1, h1);
    k_self1     <<<gNH,   B, 0, stream>>>(h1, dis, agg1);
    k_scatter1  <<<gEH,   B, 0, stream>>>(src, dst, h1, dis, agg1);
    k_relu_bias1<<<gNH,   B, 0, stream>>>(agg1, b1);
    k_gemm2     <<<gN,    B, 0, stream>>>(agg1, W2, h2);
    k_init_out  <<<gNO,   B, 0, stream>>>(h2, dis, b2, out);
    k_scatter2  <<<gE,    B, 0, stream>>>(src, dst, h2, dis, out);
}